// projection_31963146617084
// MI455X (gfx1250) — compile-verified
//
#include <hip/hip_runtime.h>

typedef float v2f __attribute__((ext_vector_type(2)));
typedef float v8f __attribute__((ext_vector_type(8)));

// ---------------------------------------------------------------------------
// Problem constants (from reference): dX (8192,64) f32, A (64,8,16,16) f32,
// out (8192,8,16,16) f32.  TAYLOR_TERMS=13, MAX_SQ=12.
// ---------------------------------------------------------------------------
#define NBATCH 8192
#define NIN    64
#define NCH    8

// ---------------------------------------------------------------------------
// Kernel 1: symplectic projection P = 0.5*(X + J X^T J), J=[[0,I],[-I,0]]
// Elementwise: P[r][c] = 0.5*(X[r][c] + sgn * X[(c+8)&15][(r+8)&15]),
//   sgn = -1 if (r<8)==(c<8) else +1.
// Output is stored permuted to the WMMA C/D wave layout:
//   Apw[((c*64 + i)*32 + lane)*8 + v] = P[i,c][M][N],
//   M = v + 8*(lane>>4), N = lane&15
// so each lane's 8 matrix elements are 32 contiguous bytes.
// ---------------------------------------------------------------------------
__global__ void sp_project_permute_kernel(const float* __restrict__ A,
                                          float* __restrict__ Apw) {
    const int tid  = blockIdx.x * blockDim.x + threadIdx.x;
    const int lane = tid & 31;
    const int w    = tid >> 5;          // one wave per (i,c) matrix, 512 total
    const int i    = w >> 3;
    const int c    = w & 7;
    const float* X = A + (size_t)(i * NCH + c) * 256;
    const int hi   = lane >> 4;
    const int N    = lane & 15;
    float* dst = Apw + ((size_t)(c * NIN + i) * 32 + lane) * 8;
#pragma unroll
    for (int v = 0; v < 8; ++v) {
        const int M   = v + 8 * hi;
        const float x  = X[M * 16 + N];
        const float xt = X[((N + 8) & 15) * 16 + ((M + 8) & 15)];
        const float sg = ((M < 8) == (N < 8)) ? -1.0f : 1.0f;
        dst[v] = 0.5f * (x + sg * xt);
    }
}

// ---------------------------------------------------------------------------
// WMMA helpers.  One wave holds a 16x16 f32 matrix in the C/D layout:
//   VGPR v, lanes 0-15:  row M=v,   col N=lane
//   VGPR v, lanes 16-31: row M=v+8, col N=lane-16
//
// 16x16x16 matmul = 4 chained V_WMMA_F32_16X16X4_F32 (K chunks of 4).
//
// B operand chunk j, vgpr t holds row K=4j+t (lanes 0-15) and K=4j+t+2
// (lanes 16-31).  Built from C/D layout with v_permlane16_swap_b32:
// swap(vdst=e[r+2], vsrc=e[r]) yields
//   vsrc_new = [e[r].lo | e[r+2].lo] = rows {r,   r+2}   (low-K chunk)
//   vdst_new = [e[r].hi | e[r+2].hi] = rows {r+8, r+10}  (high-K chunk)
// i.e. 4 swaps build all 8 B registers, no LDS traffic.
//
// A operand (16x4 per chunk: vgpr t = col 4j+t (lanes 0-15) / 4j+t+2 (16-31),
// row M = lane&15) needs a lane-ownership transpose -> per-wave LDS scratch.
// ---------------------------------------------------------------------------
__device__ __forceinline__ v8f mm16x16(const v2f aop[4], const v8f e, bool lo) {
    v2f b0, b1, b2, b3;
#if __has_builtin(__builtin_amdgcn_permlane16_swap)
    typedef unsigned int v2u __attribute__((ext_vector_type(2)));
    const v2u r02 = __builtin_amdgcn_permlane16_swap(
        __float_as_uint(e[2]), __float_as_uint(e[0]), false, false);
    const v2u r13 = __builtin_amdgcn_permlane16_swap(
        __float_as_uint(e[3]), __float_as_uint(e[1]), false, false);
    const v2u r46 = __builtin_amdgcn_permlane16_swap(
        __float_as_uint(e[6]), __float_as_uint(e[4]), false, false);
    const v2u r57 = __builtin_amdgcn_permlane16_swap(
        __float_as_uint(e[7]), __float_as_uint(e[5]), false, false);
    b0[0] = __uint_as_float(r02[1]);  b2[0] = __uint_as_float(r02[0]);
    b0[1] = __uint_as_float(r13[1]);  b2[1] = __uint_as_float(r13[0]);
    b1[0] = __uint_as_float(r46[1]);  b3[0] = __uint_as_float(r46[0]);
    b1[1] = __uint_as_float(r57[1]);  b3[1] = __uint_as_float(r57[0]);
    (void)lo;
#else
    float es[8];
#pragma unroll
    for (int v = 0; v < 8; ++v) es[v] = __shfl_xor(e[v], 16);
    b0[0] = lo ? e[0]  : es[2];  b0[1] = lo ? e[1]  : es[3];  // rows 0,2 / 1,3
    b1[0] = lo ? e[4]  : es[6];  b1[1] = lo ? e[5]  : es[7];  // rows 4,6 / 5,7
    b2[0] = lo ? es[0] : e[2];   b2[1] = lo ? es[1] : e[3];   // rows 8,10 / 9,11
    b3[0] = lo ? es[4] : e[6];   b3[1] = lo ? es[5] : e[7];   // rows 12,14 / 13,15
#endif
    v8f acc = {0.f, 0.f, 0.f, 0.f, 0.f, 0.f, 0.f, 0.f};
    acc = __builtin_amdgcn_wmma_f32_16x16x4_f32(false, aop[0], false, b0,
                                                (short)0, acc, false, false);
    acc = __builtin_amdgcn_wmma_f32_16x16x4_f32(false, aop[1], false, b1,
                                                (short)0, acc, false, false);
    acc = __builtin_amdgcn_wmma_f32_16x16x4_f32(false, aop[2], false, b2,
                                                (short)0, acc, false, false);
    acc = __builtin_amdgcn_wmma_f32_16x16x4_f32(false, aop[3], false, b3,
                                                (short)0, acc, false, false);
    return acc;
}

// spill the wave's matrix (C/D layout) row-major into its private LDS scratch
__device__ __forceinline__ void spill_rowmajor(float* scr, int N, int hi,
                                               const v8f m) {
#pragma unroll
    for (int v = 0; v < 8; ++v) scr[(v + 8 * hi) * 16 + N] = m[v];
}

// gather the A-operand chunks back from the row-major scratch
__device__ __forceinline__ void load_aop(const float* scr, int N, int hi,
                                         v2f aop[4]) {
#pragma unroll
    for (int j = 0; j < 4; ++j) {
        const float* q = scr + N * 16 + 4 * j + 2 * hi;   // 2 contiguous cols
        aop[j][0] = q[0];
        aop[j][1] = q[1];
    }
}

// ---------------------------------------------------------------------------
// Kernel 2: einsum + scaling-and-squaring expm.
// Block = 256 threads = 8 waves, c = blockIdx.y fixed per block.
// Stage Ap[:,c] (64 KB, permuted layout) in LDS once; wave w owns matrix
// (n = blockIdx.x*8 + w, c).  Dynamic LDS = 65536 bytes.
// ---------------------------------------------------------------------------
__global__ void expm_sp_kernel(const float* __restrict__ dX,
                               const float* __restrict__ Apw,
                               float* __restrict__ out) {
    extern __shared__ float lds[];            // 16384 floats
    const int c    = blockIdx.y;
    const int tid  = threadIdx.x;
    const int lane = tid & 31;
    const int wid  = __builtin_amdgcn_readfirstlane(tid >> 5);  // wave-uniform
    const int n    = blockIdx.x * 8 + wid;
    const int hi   = lane >> 4;
    const int N    = lane & 15;
    const bool lo  = (lane & 16) == 0;

    // ---- stage Ap[:,c] into LDS (straight vectorized copy) ----
    {
        const float4* src = (const float4*)(Apw + (size_t)c * NIN * 256);
        float4* dst = (float4*)lds;
#pragma unroll
        for (int k = 0; k < 16; ++k)          // 256 thr * 16 float4 = 64 KB
            dst[k * 256 + tid] = src[k * 256 + tid];
    }
    __syncthreads();

    // ---- einsum: AX[n,c] = sum_i dX[n,i] * Ap[i,c], in C/D layout ----
    const float* dxr = dX + (size_t)n * NIN;  // n uniform -> s_load path
    v8f a = {0.f, 0.f, 0.f, 0.f, 0.f, 0.f, 0.f, 0.f};
#pragma unroll
    for (int i = 0; i < NIN; ++i) {
        const float x = dxr[i];
        const float4* ap = (const float4*)(lds + (i * 32 + lane) * 8);
        const float4 p0 = ap[0], p1 = ap[1];
        a[0] += x * p0.x;  a[1] += x * p0.y;
        a[2] += x * p0.z;  a[3] += x * p0.w;
        a[4] += x * p1.x;  a[5] += x * p1.y;
        a[6] += x * p1.z;  a[7] += x * p1.w;
    }
    __syncthreads();   // Ap staging done everywhere; LDS now per-wave scratch

    // ---- induced 1-norm: max over cols of sum over rows of |a| ----
    float p = fabsf(a[0]) + fabsf(a[1]) + fabsf(a[2]) + fabsf(a[3]) +
              fabsf(a[4]) + fabsf(a[5]) + fabsf(a[6]) + fabsf(a[7]);
    p += __shfl_xor(p, 16);                   // combine row halves per column
    float nm = p;
#pragma unroll
    for (int off = 8; off >= 1; off >>= 1)    // max over 16 columns
        nm = fmaxf(nm, __shfl_xor(nm, off));

    int s = (nm > 1.0f) ? (int)ceilf(log2f(nm)) : 0;
    s = __builtin_amdgcn_readfirstlane(s);    // uniform squaring count
    const float scale = exp2f((float)(-s));

    v8f as;
#pragma unroll
    for (int v = 0; v < 8; ++v) as[v] = a[v] * scale;

    // ---- per-wave LDS scratch (1 KB) for A-operand transposes ----
    float* scr = lds + wid * 256;
    spill_rowmajor(scr, N, hi, as);
    v2f aop_s[4];
    load_aop(scr, N, hi, aop_s);              // fixed across all Taylor steps

    // ---- Horner Taylor: E = I + As/13; E = I + (As @ E)/k, k = 12..1 ----
    float Iv[8];
#pragma unroll
    for (int v = 0; v < 8; ++v) Iv[v] = (N == v + 8 * hi) ? 1.0f : 0.0f;

    v8f e;
#pragma unroll
    for (int v = 0; v < 8; ++v) e[v] = Iv[v] + as[v] * (1.0f / 13.0f);

#pragma unroll
    for (int k = 12; k >= 1; --k) {
        const v8f pr = mm16x16(aop_s, e, lo);
        const float invk = 1.0f / (float)k;   // constant-folded (unrolled)
#pragma unroll
        for (int v = 0; v < 8; ++v) e[v] = Iv[v] + pr[v] * invk;
    }

    // ---- s squarings (== reference's 12-step predicated scan) ----
    for (int it = 0; it < s; ++it) {
        spill_rowmajor(scr, N, hi, e);
        v2f aop_e[4];
        load_aop(scr, N, hi, aop_e);
        e = mm16x16(aop_e, e, lo);
    }

    // ---- store out[n][c][M][N] row-major ----
    float* o = out + ((size_t)n * NCH + c) * 256;
#pragma unroll
    for (int v = 0; v < 8; ++v) o[(v + 8 * hi) * 16 + N] = e[v];
}

// ---------------------------------------------------------------------------
extern "C" void kernel_launch(void* const* d_in, const int* in_sizes, int n_in,
                              void* d_out, int out_size, void* d_ws,
                              size_t ws_size, hipStream_t stream) {
    const float* dX = (const float*)d_in[0];   // (8192, 64)
    const float* A  = (const float*)d_in[1];   // (64, 8, 16, 16)
    float* out = (float*)d_out;                // (8192, 8, 16, 16)
    float* Apw = (float*)d_ws;                 // 512 KB permuted projection

    // 512 matrices * 32 lanes = 16384 threads
    sp_project_permute_kernel<<<64, 256, 0, stream>>>(A, Apw);

    // grid: 1024 n-tiles (8 waves each) x 8 channels; 64 KB dynamic LDS
    expm_sp_kernel<<<dim3(NBATCH / 8, NCH), 256, 64 * 1024, stream>>>(dX, Apw,
                                                                      out);
    (void)in_sizes; (void)n_in; (void)out_size; (void)ws_size;
}